// MambaBlock_36507222016766
// MI455X (gfx1250) — compile-verified
//
#include <hip/hip_runtime.h>
#include <cmath>

typedef __attribute__((ext_vector_type(2))) float v2f;
typedef __attribute__((ext_vector_type(8))) float v8f;

#define B_SZ   2
#define LSEQ   2048
#define DMODEL 1024
#define DINNER 2048
#define DTRANK 64
#define DSTATE 16
#define NXPROJ 96              // DT_RANK + 2*D_STATE
#define MROWS  (B_SZ * LSEQ)   // 4096 rows in every GEMM

// GEMM tiling
#define BM 128
#define BN 128
#define BK 16
#define GEMM_THREADS 256       // 8 waves: 2 (M) x 4 (N)

#define GLOBAL_AS __attribute__((address_space(1)))
#define LDS_AS    __attribute__((address_space(3)))

// Exact pointee type the async builtin expects (per hipcc diagnostic):
// '__attribute__((__vector_size__(4 * sizeof(int)))) int' in AS(1)/AS(3).
typedef int async_b128_t __attribute__((vector_size(16)));
#define ASYNC_GPTR(p) ((GLOBAL_AS async_b128_t*)(p))
#define ASYNC_LPTR(p) ((LDS_AS    async_b128_t*)(p))

#if __has_builtin(__builtin_amdgcn_global_load_async_to_lds_b128) && \
    __has_builtin(__builtin_amdgcn_s_wait_asynccnt)
#define HAVE_ASYNC 1
#else
#define HAVE_ASYNC 0
#endif

__device__ __forceinline__ float sigmoidf_(float v) { return 1.0f / (1.0f + expf(-v)); }

// ---------------------------------------------------------------------------
// Cooperative BMxBK (A) and BKxBN (B) tile staging, global -> LDS.
// Uses gfx1250 async-to-LDS b128 ops when available (tracked by ASYNCcnt).
// ---------------------------------------------------------------------------
__device__ __forceinline__ void stage_tiles(
    const float* __restrict__ A, const float* __restrict__ Bm,
    float* lA, float* lB, int tid, int gRow0, int gCol0, int k0,
    int lda, int ldb, int N)
{
    // A tile: BM x BK = 2048 floats = 512 x 16B chunks, 2 per thread
    #pragma unroll
    for (int c = tid; c < (BM * BK) / 4; c += GEMM_THREADS) {
        int row = c >> 2;                 // BK/4 = 4 chunks per row
        int off = (c & 3) * 4;
        const float* g = A + (size_t)(gRow0 + row) * lda + k0 + off;
        float* l = lA + row * BK + off;
#if HAVE_ASYNC
        __builtin_amdgcn_global_load_async_to_lds_b128(
            ASYNC_GPTR(g), ASYNC_LPTR(l), 0, 0);
#else
        *(float4*)l = *(const float4*)g;
#endif
    }
    // B tile: BK x BN = 2048 floats = 512 x 16B chunks, 2 per thread
    #pragma unroll
    for (int c = tid; c < (BK * BN) / 4; c += GEMM_THREADS) {
        int row = c >> 5;                 // BN/4 = 32 chunks per row
        int off = (c & 31) * 4;
        if (gCol0 + off < N) {            // guard (N=96 case); skipped cols unused
            const float* g = Bm + (size_t)(k0 + row) * ldb + gCol0 + off;
            float* l = lB + row * BN + off;
#if HAVE_ASYNC
            __builtin_amdgcn_global_load_async_to_lds_b128(
                ASYNC_GPTR(g), ASYNC_LPTR(l), 0, 0);
#else
            *(float4*)l = *(const float4*)g;
#endif
        }
    }
}

// ---------------------------------------------------------------------------
// LDS-tiled, double-buffered fp32 GEMM via V_WMMA_F32_16X16X4_F32.
// Block tile 128x128, BK=16; wave tile 64x32 (4x2 accumulators).
// epi==1: C = softplus(C + bias[col]).
// Fragment layouts per CDNA5 ISA 7.12.2 (i = lane&15, kh = lane>>4):
//   A frag:  row i,            K = {2kh, 2kh+1}  (contiguous -> ds b64)
//   B frag:  rows {2kh,2kh+1}, col i
//   C/D:     VGPR r -> row 8*kh + r, col i
// ---------------------------------------------------------------------------
__global__ __launch_bounds__(GEMM_THREADS) void wmma_gemm_f32(
    const float* __restrict__ A, const float* __restrict__ Bm,
    float* __restrict__ Cm, const float* __restrict__ bias,
    int M, int N, int K, int lda, int ldb, int ldc, int epi)
{
    __shared__ float lA[2][BM * BK];
    __shared__ float lB[2][BK * BN];

    const int tid  = threadIdx.x;
    const int lane = tid & 31;
    const int wv   = tid >> 5;
    const int i    = lane & 15;
    const int kh   = lane >> 4;
    const int row0 = (wv >> 2) * 64;     // wave M offset in block tile
    const int col0 = (wv & 3) * 32;      // wave N offset in block tile
    const int gRow0 = blockIdx.y * BM;
    const int gCol0 = blockIdx.x * BN;

    v8f acc[4][2];
    #pragma unroll
    for (int m = 0; m < 4; ++m)
        #pragma unroll
        for (int nn = 0; nn < 2; ++nn)
            acc[m][nn] = {};

    stage_tiles(A, Bm, lA[0], lB[0], tid, gRow0, gCol0, 0, lda, ldb, N);

    int cur = 0;
    for (int k0 = 0; k0 < K; k0 += BK) {
#if HAVE_ASYNC
        __builtin_amdgcn_s_wait_asynccnt(0);
#endif
        __syncthreads();                 // buf[cur] ready; buf[cur^1] free
        if (k0 + BK < K)
            stage_tiles(A, Bm, lA[cur ^ 1], lB[cur ^ 1], tid,
                        gRow0, gCol0, k0 + BK, lda, ldb, N);

        const float* As = lA[cur];
        const float* Bs = lB[cur];
        #pragma unroll
        for (int km = 0; km < BK / 4; ++km) {
            const int kb = km * 4 + 2 * kh;
            v2f a[4], b[2];
            #pragma unroll
            for (int m = 0; m < 4; ++m)
                a[m] = *(const v2f*)&As[(row0 + m * 16 + i) * BK + kb];
            #pragma unroll
            for (int nn = 0; nn < 2; ++nn) {
                b[nn][0] = Bs[kb * BN + col0 + nn * 16 + i];
                b[nn][1] = Bs[(kb + 1) * BN + col0 + nn * 16 + i];
            }
            #pragma unroll
            for (int m = 0; m < 4; ++m)
                #pragma unroll
                for (int nn = 0; nn < 2; ++nn)
                    acc[m][nn] = __builtin_amdgcn_wmma_f32_16x16x4_f32(
                        false, a[m], false, b[nn], (short)0, acc[m][nn],
                        false, false);
        }
        cur ^= 1;
    }

    #pragma unroll
    for (int m = 0; m < 4; ++m) {
        #pragma unroll
        for (int nn = 0; nn < 2; ++nn) {
            const int col = gCol0 + col0 + nn * 16 + i;
            if (col < N) {
                const float bv = (epi == 1) ? bias[col] : 0.0f;
                #pragma unroll
                for (int r = 0; r < 8; ++r) {
                    const int row = gRow0 + row0 + m * 16 + kh * 8 + r;
                    float v = acc[m][nn][r];
                    if (epi == 1) {      // bias + softplus
                        v += bv;
                        v = (v > 20.0f) ? v : log1pf(expf(v));
                    }
                    Cm[(size_t)row * ldc + col] = v;
                }
            }
        }
    }
}

// ---------------------------------------------------------------------------
// Depthwise causal conv (D_CONV=4) + SiLU.  Reads x_in_raw = xz[:, :DINNER].
// ---------------------------------------------------------------------------
__global__ __launch_bounds__(256) void conv_silu_kernel(
    const float* __restrict__ xz, const float* __restrict__ Wc,
    const float* __restrict__ bc, float* __restrict__ xin, int total)
{
    int idx = blockIdx.x * blockDim.x + threadIdx.x;
    if (idx >= total) return;
    int d  = idx & (DINNER - 1);
    int bt = idx >> 11;
    int t  = bt & (LSEQ - 1);
    int b  = bt >> 11;

    float acc = bc[d];
    #pragma unroll
    for (int k = 0; k < 4; ++k) {
        int ts = t - 3 + k;
        if (ts >= 0)
            acc += xz[(size_t)(b * LSEQ + ts) * (2 * DINNER) + d] * Wc[k * DINNER + d];
    }
    xin[idx] = acc * sigmoidf_(acc);
}

// ---------------------------------------------------------------------------
// Selective scan: one lane per (b, d, n); 16-lane shfl_xor reduction over n.
// Writes y * silu(z) in place over the dt buffer.
// ---------------------------------------------------------------------------
__global__ __launch_bounds__(256) void scan_kernel(
    const float* __restrict__ xin, float* __restrict__ dt_y,
    const float* __restrict__ dbc, const float* __restrict__ xz,
    const float* __restrict__ A_log, const float* __restrict__ Dvec)
{
    int gtid = blockIdx.x * blockDim.x + threadIdx.x;
    int n  = gtid & 15;
    int ch = gtid >> 4;
    if (ch >= B_SZ * DINNER) return;
    int d = ch & (DINNER - 1);
    int b = ch >> 11;

    const float Ad = -expf(A_log[d * DSTATE + n]);
    const float Dd = Dvec[d];
    float state = 0.0f;

    for (int t = 0; t < LSEQ; ++t) {
        size_t base = (size_t)(b * LSEQ + t);
        float xv  = xin[base * DINNER + d];
        float dtv = dt_y[base * DINNER + d];
        float Bv  = dbc[base * NXPROJ + DTRANK + n];
        float Cv  = dbc[base * NXPROJ + DTRANK + DSTATE + n];

        float dA = expf(fminf(fmaxf(dtv * Ad, -10.0f), 10.0f));
        state = fminf(fmaxf(state * dA + xv * dtv * Bv, -10.0f), 10.0f);

        float yp = state * Cv;
        yp += __shfl_xor(yp, 8, 32);
        yp += __shfl_xor(yp, 4, 32);
        yp += __shfl_xor(yp, 2, 32);
        yp += __shfl_xor(yp, 1, 32);

        if (n == 0) {
            float y  = yp + Dd * xv;
            float zv = xz[base * (2 * DINNER) + DINNER + d];
            dt_y[base * DINNER + d] = y * (zv * sigmoidf_(zv));
        }
    }
}

// ---------------------------------------------------------------------------
extern "C" void kernel_launch(void* const* d_in, const int* in_sizes, int n_in,
                              void* d_out, int out_size, void* d_ws, size_t ws_size,
                              hipStream_t stream)
{
    const float* x       = (const float*)d_in[0];
    const float* W_in    = (const float*)d_in[1];
    const float* W_conv  = (const float*)d_in[2];
    const float* b_conv  = (const float*)d_in[3];
    const float* W_xproj = (const float*)d_in[4];
    const float* W_dt    = (const float*)d_in[5];
    const float* b_dt    = (const float*)d_in[6];
    const float* A_log   = (const float*)d_in[7];
    const float* Dvec    = (const float*)d_in[8];
    const float* W_out   = (const float*)d_in[9];
    float* out = (float*)d_out;

    // workspace layout (floats)
    float* ws  = (float*)d_ws;
    float* xz  = ws;                                      // MROWS x 4096
    float* xin = xz  + (size_t)MROWS * 2 * DINNER;        // MROWS x 2048
    float* dbc = xin + (size_t)MROWS * DINNER;            // MROWS x 96
    float* dtb = dbc + (size_t)MROWS * NXPROJ;            // MROWS x 2048 (dt, then y*silu(z))

    dim3 blk(GEMM_THREADS);

    // 1) xz = x @ W_in                    (4096x1024 @ 1024x4096)
    wmma_gemm_f32<<<dim3((2 * DINNER + BN - 1) / BN, MROWS / BM), blk, 0, stream>>>(
        x, W_in, xz, nullptr, MROWS, 2 * DINNER, DMODEL, DMODEL, 2 * DINNER, 2 * DINNER, 0);

    // 2) depthwise conv + SiLU -> xin
    int total = MROWS * DINNER;
    conv_silu_kernel<<<(total + 255) / 256, 256, 0, stream>>>(xz, W_conv, b_conv, xin, total);

    // 3) dbc = xin @ W_xproj              (4096x2048 @ 2048x96)
    wmma_gemm_f32<<<dim3((NXPROJ + BN - 1) / BN, MROWS / BM), blk, 0, stream>>>(
        xin, W_xproj, dbc, nullptr, MROWS, NXPROJ, DINNER, DINNER, NXPROJ, NXPROJ, 0);

    // 4) dt = softplus(dbc[:, :64] @ W_dt + b_dt)   (K=64, N=2048)
    wmma_gemm_f32<<<dim3(DINNER / BN, MROWS / BM), blk, 0, stream>>>(
        dbc, W_dt, dtb, b_dt, MROWS, DINNER, DTRANK, NXPROJ, DINNER, DINNER, 1);

    // 5) selective scan + SiLU(z) gate, writes into dtb
    scan_kernel<<<(B_SZ * DINNER * 16) / 256, 256, 0, stream>>>(
        xin, dtb, dbc, xz, A_log, Dvec);

    // 6) out = y @ W_out                  (4096x2048 @ 2048x1024)
    wmma_gemm_f32<<<dim3(DMODEL / BN, MROWS / BM), blk, 0, stream>>>(
        dtb, W_out, out, nullptr, MROWS, DMODEL, DINNER, DINNER, DMODEL, DMODEL, 0);
}